// CodeBP_29265907155195
// MI455X (gfx1250) — compile-verified
//
#include <hip/hip_runtime.h>
#include <math.h>

#define TPB 256
#define MAXBLK 8192

typedef float v2f __attribute__((ext_vector_type(2)));

static __device__ __forceinline__ float nt_loadf(const float* p) {
    return __builtin_nontemporal_load(p);
}
static __device__ __forceinline__ int nt_loadi(const int* p) {
    return __builtin_nontemporal_load(p);
}
static __device__ __forceinline__ v2f nt_loadf2(const float* p) {
    return __builtin_nontemporal_load((const v2f*)p);
}

// Raw LDS byte offset of a __shared__ object (generic -> addrspace(3) -> int).
typedef __attribute__((address_space(3))) const void lds_cvoid;
static __device__ __forceinline__ unsigned lds_addr_of(const float* p) {
    return (unsigned)(size_t)(lds_cvoid*)p;
}

// gfx1250 async copy: global -> LDS, tracked by ASYNCcnt (ISA 15.18.3 op 96).
static __device__ __forceinline__ void async_load_to_lds_b32(const float* g, unsigned lds) {
    asm volatile("global_load_async_to_lds_b32 %0, %1, off"
                 :: "v"(lds), "v"(g) : "memory");
}
static __device__ __forceinline__ void wait_asynccnt0() {
    asm volatile("s_wait_asynccnt 0x0" ::: "memory");
}

// ---------------------------------------------------------------------------
// Kernel 1: variable -> factor messages.
//   prior = 0.5*(log m0 - log m1) + 0.5*(log p0 - log p1)
//   Hsx per edge with the inf-case logic, then t = tanh(Hsx),
//   scattered through perm into factor-side SoA: t_fac[k*K + a], e_f = 6a+k.
// Streamed-once inputs use non-temporal loads (preserve L2 for edge scratch).
// ---------------------------------------------------------------------------
__global__ void bp_var2fac(const float* __restrict__ ps,
                           const float* __restrict__ Min,
                           const float* __restrict__ Hxs_prev,
                           const int*   __restrict__ perm,
                           float*       __restrict__ t_fac,
                           int N, int K)
{
    const int stride = gridDim.x * blockDim.x;
    for (int i = blockIdx.x * blockDim.x + threadIdx.x; i < N; i += stride) {
        const int ip = i + stride;              // prefetch next grid-stride chunk
        if (ip < N) {
            __builtin_prefetch(&ps[2 * ip], 0, 0);
            __builtin_prefetch(&Min[2 * ip], 0, 0);
            __builtin_prefetch(&Hxs_prev[3 * ip], 0, 0);
            __builtin_prefetch(&perm[3 * ip], 0, 0);
        }

        const v2f p = nt_loadf2(ps + 2 * i);
        const v2f m = nt_loadf2(Min + 2 * i);
        const float prior = 0.5f * (logf(m.x) - logf(m.y))
                          + 0.5f * (logf(p.x) - logf(p.y));

        float h[3];
        h[0] = nt_loadf(Hxs_prev + 3 * i + 0);
        h[1] = nt_loadf(Hxs_prev + 3 * i + 1);
        h[2] = nt_loadf(Hxs_prev + 3 * i + 2);

        const bool inf0 = __builtin_isinf(h[0]);
        const bool inf1 = __builtin_isinf(h[1]);
        const bool inf2 = __builtin_isinf(h[2]);
        const int  sure = (int)inf0 + (int)inf1 + (int)inf2;
        const float ll_sum   = (inf0 ? 0.0f : h[0]) + (inf1 ? 0.0f : h[1]) + (inf2 ? 0.0f : h[2]);
        const float sure_sum = (inf0 ? h[0] : 0.0f) + (inf1 ? h[1] : 0.0f) + (inf2 ? h[2] : 0.0f);
        const bool  inf_chk[3] = {inf0, inf1, inf2};

#pragma unroll
        for (int j = 0; j < 3; ++j) {
            float hsx;
            if (sure == 0) {
                hsx = prior + ll_sum - h[j];                    // extrinsic sum
            } else if (sure == 1) {
                hsx = inf_chk[j] ? (prior + ll_sum) : sure_sum; // inf slot gets finite
            } else {
                hsx = __builtin_isnan(sure_sum) ? 0.0f : (prior + sure_sum);
            }
            const float t = tanhf(hsx);                         // TRANS v_tanh_f32
            const int dst = nt_loadi(perm + 3 * i + j);         // factor-side flat edge
            const int a = dst / 6;
            const int k = dst - 6 * a;
            t_fac[k * K + a] = t;   // RT store: keep edge scratch resident in 192MB L2
        }
    }
}

// ---------------------------------------------------------------------------
// Kernel 2: factor -> variable messages.
//   Six coalesced t_fac streams staged lane-private into LDS with the gfx1250
//   async copy engine (ASYNCcnt); overlapped with inv_perm/x loads. No barrier:
//   each thread reads only the LDS slots its own async loads wrote.
//   prod_nz = sgn * prod(nonzero t); null = #zeros; branch per reference.
//   Scatter through inv_perm into variable-side SoA: h_var[j*N + i], e_v = 3i+j.
// ---------------------------------------------------------------------------
__global__ void bp_fac2var(const float* __restrict__ t_fac,
                           const int*   __restrict__ x,
                           const int*   __restrict__ inv_perm,
                           float*       __restrict__ h_var,
                           int N, int K)
{
    __shared__ float sh_t[6 * TPB];
    const int tid = threadIdx.x;
    const int stride = gridDim.x * blockDim.x;
    for (int a = blockIdx.x * blockDim.x + tid; a < K; a += stride) {
        const int ap = a + stride;
        if (ap < K) {
            __builtin_prefetch(&x[ap], 0, 0);
            __builtin_prefetch(&inv_perm[6 * ap], 0, 0);
        }

        // Issue the six async global->LDS copies for this lane's check inputs.
#pragma unroll
        for (int k = 0; k < 6; ++k) {
            async_load_to_lds_b32(t_fac + k * K + a,
                                  lds_addr_of(&sh_t[k * TPB + tid]));
        }

        // Independent work while the async engine runs.
        const float sgn = 1.0f - 2.0f * (float)nt_loadi(x + a);
        int v[6];
#pragma unroll
        for (int k = 0; k < 6; ++k) v[k] = nt_loadi(inv_perm + 6 * a + k);

        wait_asynccnt0();                      // s_wait_asynccnt 0

        float t[6];
#pragma unroll
        for (int k = 0; k < 6; ++k) t[k] = sh_t[k * TPB + tid];

        int nzero = 0;
        float prod = sgn;
        bool zero[6];
#pragma unroll
        for (int k = 0; k < 6; ++k) {
            zero[k] = (t[k] == 0.0f);
            nzero += (int)zero[k];
            prod = zero[k] ? prod : prod * t[k];               // prod_nz
        }

        float out[6];
        if (nzero == 0) {
#pragma unroll
            for (int k = 0; k < 6; ++k) out[k] = atanhf(prod / t[k]);
        } else if (nzero == 1) {
#pragma unroll
            for (int k = 0; k < 6; ++k) out[k] = zero[k] ? atanhf(prod) : 0.0f;
        } else {
#pragma unroll
            for (int k = 0; k < 6; ++k) out[k] = 0.0f;
        }

#pragma unroll
        for (int k = 0; k < 6; ++k) {
            const int i = v[k] / 3;
            const int j = v[k] - 3 * i;
            h_var[j * N + i] = out[k];  // RT store: stays in L2 for kernel 3
        }
    }
}

// ---------------------------------------------------------------------------
// Kernel 3: output beliefs. 3 coalesced loads per variable, tanh + NaN/inf fix,
// non-temporal store of M_out = [0.5+0.5d, 0.5-0.5d].
// ---------------------------------------------------------------------------
__global__ void bp_out(const float* __restrict__ h_var,
                       float*       __restrict__ M_out,
                       int N)
{
    const int stride = gridDim.x * blockDim.x;
    for (int i = blockIdx.x * blockDim.x + threadIdx.x; i < N; i += stride) {
        const float h0 = h_var[i];
        const float h1 = h_var[N + i];
        const float h2 = h_var[2 * N + i];

        float d = tanhf(h0 + h1 + h2);
        if (__builtin_isnan(d)) {
            const bool i0 = __builtin_isinf(h0);
            const bool i1 = __builtin_isinf(h1);
            const bool i2 = __builtin_isinf(h2);
            if (i0 | i1 | i2) {
                d = tanhf((i0 ? 0.0f : h0) + (i1 ? 0.0f : h1) + (i2 ? 0.0f : h2));
            }
        }
        __builtin_nontemporal_store(0.5f + 0.5f * d, M_out + 2 * i);
        __builtin_nontemporal_store(0.5f - 0.5f * d, M_out + 2 * i + 1);
    }
}

static inline int grid_for(int n) {
    int b = (n + TPB - 1) / TPB;
    return b < MAXBLK ? b : MAXBLK;
}

extern "C" void kernel_launch(void* const* d_in, const int* in_sizes, int n_in,
                              void* d_out, int out_size, void* d_ws, size_t ws_size,
                              hipStream_t stream) {
    const float* ps       = (const float*)d_in[0];  // [N,2]
    const float* Min      = (const float*)d_in[1];  // [N,2]
    const float* Hxs_prev = (const float*)d_in[2];  // [N,3]
    const int*   x        = (const int*)  d_in[3];  // [K,1]
    const int*   perm     = (const int*)  d_in[4];  // [E]
    const int*   inv_perm = (const int*)  d_in[5];  // [E]

    const int N = in_sizes[0] / 2;
    const int K = in_sizes[3];
    const int E = in_sizes[4];          // = 3N = 6K

    float* t_fac = (float*)d_ws;        // E floats: tanh(Hsx) in factor-SoA order
    float* h_var = t_fac + E;           // E floats: Hxs in variable-SoA order
    float* out   = (float*)d_out;       // [N,2]

    bp_var2fac<<<grid_for(N), TPB, 0, stream>>>(ps, Min, Hxs_prev, perm, t_fac, N, K);
    bp_fac2var<<<grid_for(K), TPB, 0, stream>>>(t_fac, x, inv_perm, h_var, N, K);
    bp_out    <<<grid_for(N), TPB, 0, stream>>>(h_var, out, N);
}